// LocallyConnected3_38886633898703
// MI455X (gfx1250) — compile-verified
//
#include <hip/hip_runtime.h>
#include <stdint.h>

// Problem constants (from reference setup_inputs)
#define B_TOT   128
#define C_IN    32
#define C_OUT   8
#define S_LEN   8192
#define S_TILE  32      // s positions per workgroup
#define B_TILE  16      // batch rows per workgroup (= WMMA M)

typedef float    v2f __attribute__((ext_vector_type(2)));
typedef float    v8f __attribute__((ext_vector_type(8)));
typedef uint32_t v4u __attribute__((ext_vector_type(4)));
typedef int32_t  v4i __attribute__((ext_vector_type(4)));
typedef int32_t  v8i __attribute__((ext_vector_type(8)));

// ---- LDS layout (bytes). x/w2 tiles use TDM padding: +1 DWORD every 256 so the
// per-lane gathers (stride 1024 dwords -> 1028 dwords) spread across banks (4*lane).
#define X_OFF    0u
#define X_SZ     65792u                 // 16384 dwords + 64 pad dwords
#define W2_OFF   (X_OFF + X_SZ)         // 65792
#define W2_SZ    32896u                 // 8192 dwords + 32 pad dwords
#define W1S_OFF  (W2_OFF + W2_SZ)       // 98688
#define W1S_SZ   4096u
#define BIAS_OFF (W1S_OFF + W1S_SZ)     // 102784
#define BIAS_SZ  1024u
#define OUT_OFF  (BIAS_OFF + BIAS_SZ)   // 103808, row stride 33 dwords (bank spread)
#define OUT_SZ   (128u * 33u * 4u)      // 16896
#define LDS_TOTAL (OUT_OFF + OUT_SZ)    // 120704 bytes -> 2 WGs / WGP

// padded linear index for TDM pad_interval=256dw, pad_amount=1dw
__device__ __forceinline__ int padded(int n) { return n + (n >> 8); }

// gfx1250 hardware tanh (transcendental unit); chain of fallbacks.
__device__ __forceinline__ float fast_tanh(float v) {
#if __has_builtin(__builtin_amdgcn_tanhf)
  return __builtin_amdgcn_tanhf(v);
#elif __has_builtin(__builtin_amdgcn_tanh_f32)
  return __builtin_amdgcn_tanh_f32(v);
#else
  float r;
  asm volatile("v_tanh_f32 %0, %1\n\tv_nop" : "=v"(r) : "v"(v));
  return r;
#endif
}

// ---- Tensor Data Mover: build a D# (groups 0..3) and issue tensor_load_to_lds.
// Tile dims: dim0 = contiguous s (fastest), dim1/dim2 strided rows/slabs.
// This toolchain's builtin takes 6 args: (v4u, v8i, v4i, v4i, v8i, i32 cpol);
// the 4 architectural D# groups are 4+8+4+4 dwords, the extra v8i is zero-filled
// (unused with count=1 and gather disabled).
__device__ __forceinline__ void tdm_load_tile(
    uint32_t lds_addr, const void* gptr,
    uint32_t td0, uint32_t td1, uint32_t td2,
    uint32_t tl0, uint32_t tl1, uint32_t tl2,
    uint64_t st0, uint64_t st1, uint64_t st2,
    bool pad)
{
#if __has_builtin(__builtin_amdgcn_tensor_load_to_lds)
  uint64_t ga = (uint64_t)(uintptr_t)gptr;
  v4u g0;
  g0[0] = 1u;                                       // count=1, user descriptor
  g0[1] = lds_addr;                                 // LDS byte address
  g0[2] = (uint32_t)ga;                             // global_addr[31:0]
  g0[3] = (uint32_t)((ga >> 32) & 0x1FFFFFFu)       // global_addr[56:32]
        | (2u << 30);                               // type = 2 (image)

  uint32_t flags = (2u << 16);                      // data_size = 4B; wg_mask=0
  if (pad) flags |= (1u << 20)                      // pad_enable
                 |  (7u << 22)                      // pad_interval: 256 dwords
                 |  (0u << 25);                     // pad_amount: 1 dword
  v8i g1;
  g1[0] = (int32_t)flags;
  g1[1] = (int32_t)((td0 & 0xFFFFu) << 16);                          // tdim0 lo
  g1[2] = (int32_t)((td0 >> 16) | ((td1 & 0xFFFFu) << 16));
  g1[3] = (int32_t)((td1 >> 16) | (tl0 << 16));                      // tile_dim0
  g1[4] = (int32_t)(tl1 | (tl2 << 16));                              // tile_dim1/2
  g1[5] = (int32_t)(uint32_t)st0;                                    // stride0 lo
  g1[6] = (int32_t)((uint32_t)(st0 >> 32) | ((uint32_t)(st1 & 0xFFFFu) << 16));
  g1[7] = (int32_t)(uint32_t)(st1 >> 16);                            // stride1 hi
  v4i g2;
  g2[0] = (int32_t)td2;                                              // tensor_dim2
  g2[1] = 0;                                                         // tensor_dim3
  g2[2] = (int32_t)(uint32_t)st2;                                    // stride2 lo
  g2[3] = (int32_t)((uint32_t)(st2 >> 32) & 0xFFFFu);                // tile_dim3=0
  v4i g3 = {0, 0, 0, 0};
  v8i g4 = {0, 0, 0, 0, 0, 0, 0, 0};                // extra group (clang-23 form)
  __builtin_amdgcn_tensor_load_to_lds(g0, g1, g2, g3, g4, 0);
#else
  (void)lds_addr; (void)gptr; (void)td0; (void)td1; (void)td2;
  (void)tl0; (void)tl1; (void)tl2; (void)st0; (void)st1; (void)st2; (void)pad;
#endif
}

// ---- Pass 1: w1sum[j,s] = sum_i w1[i,j,s]  (32 MB -> 1 MB, float4 coalesced)
__global__ void __launch_bounds__(256) lc3_w1sum(const float* __restrict__ w1,
                                                 float* __restrict__ w1sum)
{
  const int t = blockIdx.x * 256 + threadIdx.x;       // float4 index over j*s
  const float4* w1v = (const float4*)w1;
  float4 acc = make_float4(0.f, 0.f, 0.f, 0.f);
#pragma unroll
  for (int i = 0; i < C_IN; ++i) {
    float4 v = w1v[(size_t)i * (C_IN * S_LEN / 4) + t];
    acc.x += v.x; acc.y += v.y; acc.z += v.z; acc.w += v.w;
  }
  ((float4*)w1sum)[t] = acc;
}

// ---- Pass 2: per s: out(16b x 8o) = tanh( tanh(x*w1sum) @ w2^T + bias )
// WG = 256 threads (8 waves). Wave w handles s_loc = 4w..4w+3, each as a
// 16x16x(8x4) chained V_WMMA_F32_16X16X4_F32 with C preloaded with bias.
__global__ void __launch_bounds__(256) lc3_main(
    const float* __restrict__ x, const float* __restrict__ w2,
    const float* __restrict__ bias, const float* __restrict__ w1sum,
    float* __restrict__ out)
{
  extern __shared__ uint8_t smem[];
  const int s0 = blockIdx.x * S_TILE;
  const int b0 = blockIdx.y * B_TILE;
  const int wv = threadIdx.x >> 5;

#if __has_builtin(__builtin_amdgcn_tensor_load_to_lds)
  if (wv == 0) {
    const uint32_t lds_base = (uint32_t)(uintptr_t)(void*)smem;  // low32 = LDS offset
    // x tile: 3D (s:32 contig, j:32 stride S, b:16 stride 32*S), padded
    tdm_load_tile(lds_base + X_OFF,
                  x + (uint64_t)b0 * C_IN * S_LEN + s0,
                  S_TILE, C_IN, B_TILE,  S_TILE, C_IN, B_TILE,
                  (uint64_t)S_LEN, (uint64_t)C_IN * S_LEN, 0, true);
    // w2 tile: 2D (s:32, rows o*32+j : 256, stride S), padded
    tdm_load_tile(lds_base + W2_OFF, w2 + s0,
                  S_TILE, C_OUT * C_IN, 0,  S_TILE, C_OUT * C_IN, 0,
                  (uint64_t)S_LEN, 0, 0, true);
    // w1sum tile: 2D (s:32, j:32) — read broadcast, no pad needed
    tdm_load_tile(lds_base + W1S_OFF, w1sum + s0,
                  S_TILE, C_IN, 0,  S_TILE, C_IN, 0, (uint64_t)S_LEN, 0, 0, false);
    // bias tile: 2D (s:32, o:8)
    tdm_load_tile(lds_base + BIAS_OFF, bias + s0,
                  S_TILE, C_OUT, 0,  S_TILE, C_OUT, 0, (uint64_t)S_LEN, 0, 0, false);
#if __has_builtin(__builtin_amdgcn_s_wait_tensorcnt)
    __builtin_amdgcn_s_wait_tensorcnt(0);
#endif
  }
#else
  // Cooperative fallback (same padded layout) if TDM builtin is unavailable.
  for (int n = threadIdx.x; n < B_TILE * C_IN * S_TILE; n += 256) {
    int s = n & 31, j = (n >> 5) & 31, bl = n >> 10;
    ((float*)(smem + X_OFF))[padded(n)] =
        x[((uint64_t)(b0 + bl) * C_IN + j) * S_LEN + s0 + s];
  }
  for (int n = threadIdx.x; n < C_OUT * C_IN * S_TILE; n += 256) {
    int s = n & 31, r = n >> 5;
    ((float*)(smem + W2_OFF))[padded(n)] = w2[(uint64_t)r * S_LEN + s0 + s];
  }
  for (int n = threadIdx.x; n < C_IN * S_TILE; n += 256) {
    int s = n & 31, j = n >> 5;
    ((float*)(smem + W1S_OFF))[n] = w1sum[(uint64_t)j * S_LEN + s0 + s];
  }
  for (int n = threadIdx.x; n < C_OUT * S_TILE; n += 256) {
    int s = n & 31, o = n >> 5;
    ((float*)(smem + BIAS_OFF))[n] = bias[(uint64_t)o * S_LEN + s0 + s];
  }
#endif
  __syncthreads();

  const int lane = threadIdx.x & 31;
  const int half = lane >> 4;       // K half-select for A/B, M half for D
  const int lm   = lane & 15;       // A: M(=local b); B/D: N(=o)
  float* xs  = (float*)(smem + X_OFF);
  float* w2s = (float*)(smem + W2_OFF);
  float* gs  = (float*)(smem + W1S_OFF);
  float* bbs = (float*)(smem + BIAS_OFF);
  float* os  = (float*)(smem + OUT_OFF);

  for (int i = 0; i < 4; ++i) {
    const int s = wv * 4 + i;
    // C init: all 8 C VGPRs of lane l hold bias[o = l%16] (cols 8..15 unused)
    float binit = bbs[(lm & 7) * S_TILE + s];
    if (lm >= C_OUT) binit = 0.0f;
    v8f c = {binit, binit, binit, binit, binit, binit, binit, binit};
#pragma unroll
    for (int kb = 0; kb < 8; ++kb) {
      const int j0 = kb * 4 + half * 2;       // K = j0, j0+1 for comps x,y
      v2f a, b;
      // A: h[b0+lm, j, s] = tanh(x * w1sum)   (banks = 4*lm: conflict-free-ish)
      int nx0 = (lm * C_IN + j0) * S_TILE + s;
      int nx1 = nx0 + S_TILE;
      float xv0 = xs[padded(nx0)];
      float xv1 = xs[padded(nx1)];
      a.x = fast_tanh(xv0 * gs[j0 * S_TILE + s]);
      a.y = fast_tanh(xv1 * gs[(j0 + 1) * S_TILE + s]);
      // B: w2[o=lm, j, s]  (lanes 8..15 clamp to o-8; their D cols are discarded)
      int nw0 = ((lm & 7) * C_IN + j0) * S_TILE + s;
      int nw1 = nw0 + S_TILE;
      b.x = w2s[padded(nw0)];
      b.y = w2s[padded(nw1)];
      // D = A(16x4) x B(4x16) + C, full f32
      c = __builtin_amdgcn_wmma_f32_16x16x4_f32(false, a, false, b,
                                                (short)0, c, false, false);
    }
    // D -> LDS (row stride 33 dwords for bank spread); o = lm, M = v + 8*half
    if (lm < C_OUT) {
#pragma unroll
      for (int v = 0; v < 8; ++v) {
        const int bl = v + half * 8;
        os[(bl * C_OUT + lm) * 33 + s] = fast_tanh(c[v]);
      }
    }
  }
  __syncthreads();

  // Coalesced 128B-line stores along s: row r = bl*8+o  ->  out[b0*8 + r][s0+s]
  for (int idx = threadIdx.x; idx < B_TILE * C_OUT * S_TILE; idx += 256) {
    const int r = idx >> 5;
    const int s = idx & 31;
    out[((uint64_t)(b0 * C_OUT + r)) * S_LEN + s0 + s] = os[r * 33 + s];
  }
}

extern "C" void kernel_launch(void* const* d_in, const int* in_sizes, int n_in,
                              void* d_out, int out_size, void* d_ws, size_t ws_size,
                              hipStream_t stream) {
  (void)in_sizes; (void)n_in; (void)out_size; (void)ws_size;
  const float* x    = (const float*)d_in[0];
  const float* w1   = (const float*)d_in[1];
  const float* w2   = (const float*)d_in[2];
  const float* bias = (const float*)d_in[3];
  float* out   = (float*)d_out;
  float* w1sum = (float*)d_ws;                    // 1 MB scratch

  lc3_w1sum<<<dim3((C_IN * S_LEN / 4) / 256), dim3(256), 0, stream>>>(w1, w1sum);
  lc3_main<<<dim3(S_LEN / S_TILE, B_TOT / B_TILE), dim3(256), LDS_TOTAL, stream>>>(
      x, w2, bias, w1sum, out);
}